// ldl_loss_15805479649531
// MI455X (gfx1250) — compile-verified
//
#include <hip/hip_runtime.h>
#include <math.h>

typedef float v2f __attribute__((ext_vector_type(2)));
typedef float v8f __attribute__((ext_vector_type(8)));

#define BATCH 32
#define CH 3
#define H 512
#define W 512
#define HW (H * W)            // 262144
#define KS 7
#define PAD 3
#define TILE 16
#define HALO (TILE + KS - 1)  // 22
#define KPAD 24               // K padded to multiple of 4 for 16x16x4 WMMA
#define RROWS 24              // R tile rows incl. zero pad
#define RCOLS 32              // R tile cols incl. zero pad
#define RSTRIDE 41            // odd + 2*41%64=18: conflict-free for B pattern
#define TSTRIDE 37            // odd + 8*37%64=40: conflict-free T store/load
#define WAVES 8
#define TILES_PER_IMG ((H / TILE) * (W / TILE))   // 1024
#define NTILES (BATCH * TILES_PER_IMG)            // 32768

__device__ __forceinline__ int reflect_idx(int p) {
    p = (p < 0) ? -p : p;                        // jnp reflect: -1 -> 1
    return (p > H - 1) ? (2 * (H - 1) - p) : p;  // 512 -> 510
}

__device__ __forceinline__ float wave_sum32(float x) {
#pragma unroll
    for (int m = 16; m >= 1; m >>= 1) x += __shfl_xor(x, m, 32);
    return x;
}

// ---------------------------------------------------------------------------
// Kernel 0: zero the 96 accumulator floats (sumB, sumsqB, Sb)
// ---------------------------------------------------------------------------
__global__ void ldl_init_kernel(float* acc) {
    if (threadIdx.x < 96) acc[threadIdx.x] = 0.0f;
}

// ---------------------------------------------------------------------------
// Kernel 1: residual = sum_c |gt - net|, store to ws; accumulate per-batch
// sum / sumsq for the patch-level variance. One pixel per thread.
// 201 MB streamed read (HBM-bound), 33.5 MB write (L2-resident afterwards).
// ---------------------------------------------------------------------------
__global__ __launch_bounds__(256) void ldl_residual_kernel(
    const float* __restrict__ net, const float* __restrict__ gt,
    float* __restrict__ res, float* __restrict__ sumB, float* __restrict__ sumsqB) {
    const int b = blockIdx.y;
    const int pix = blockIdx.x * blockDim.x + threadIdx.x;  // 0..HW-1
    const size_t base = (size_t)b * CH * HW + pix;
    float r = fabsf(gt[base] - net[base]) +
              fabsf(gt[base + HW] - net[base + HW]) +
              fabsf(gt[base + 2 * HW] - net[base + 2 * HW]);
    res[(size_t)b * HW + pix] = r;

    float s1 = wave_sum32(r);
    float s2 = wave_sum32(r * r);
    __shared__ float ls1[WAVES], ls2[WAVES];
    const int wave = threadIdx.x >> 5, lane = threadIdx.x & 31;
    if (lane == 0) { ls1[wave] = s1; ls2[wave] = s2; }
    __syncthreads();
    if (threadIdx.x == 0) {
        float a = 0.f, c = 0.f;
#pragma unroll
        for (int i = 0; i < WAVES; ++i) { a += ls1[i]; c += ls2[i]; }
        atomicAdd(&sumB[b], a);
        atomicAdd(&sumsqB[b], c);
    }
}

// ---------------------------------------------------------------------------
// 7x7 box sum of a zero-padded 24x32 halo tile R as two banded GEMMs on the
// matrix pipes:
//   T(16x24) = Mv(16x24) * R(24x24)     (vertical 7-sum)
//   S(16x16) = T(16x24)  * Mh(24x16)    (horizontal 7-sum)
// with V_WMMA_F32_16X16X4_F32 (6 K-chunks each). Band matrices Mv/Mh are
// synthesized per-lane from lane-id compares (A 16x4 layout: lanes 0-15 hold
// K={0,1}, lanes 16-31 hold K={2,3}; B 4x16 layout: low lanes rows {k,k+1},
// high lanes rows {k+2,k+3}). All DS accesses are unconditional: padding
// lives in LDS, so no exec-mask divergence around ds_load/ds_store.
// ---------------------------------------------------------------------------
__device__ __forceinline__ v8f box7x7_wmma(const float* R, float* T,
                                           bool squared, int lane) {
    const int halfM = lane & 15;
    const int kOff = (lane < 16) ? 0 : 2;
    v8f accLo = {};  // T columns 0..15
    v8f accHi = {};  // T columns 16..23 (22..23 are zero)
#pragma unroll
    for (int q = 0; q < 6; ++q) {
        const int k0 = 4 * q + kOff;
        // A = Mv chunk: Mv[m][i] = 1 iff m <= i <= m+6  (0 for i >= 22)
        v2f a;
        a.x = (k0 >= halfM && k0 <= halfM + 6) ? 1.0f : 0.0f;
        a.y = (k0 + 1 >= halfM && k0 + 1 <= halfM + 6) ? 1.0f : 0.0f;
        // B = R rows {k0, k0+1} (this lane half), cols 0..15 and 16..31
        float r0 = R[k0 * RSTRIDE + halfM];
        float r1 = R[(k0 + 1) * RSTRIDE + halfM];
        float h0 = R[k0 * RSTRIDE + 16 + halfM];
        float h1 = R[(k0 + 1) * RSTRIDE + 16 + halfM];
        if (squared) { r0 *= r0; r1 *= r1; h0 *= h0; h1 *= h1; }
        v2f blo; blo.x = r0; blo.y = r1;
        v2f bhi; bhi.x = h0; bhi.y = h1;
        accLo = __builtin_amdgcn_wmma_f32_16x16x4_f32(false, a, false, blo,
                                                      (short)0, accLo, false, false);
        accHi = __builtin_amdgcn_wmma_f32_16x16x4_f32(false, a, false, bhi,
                                                      (short)0, accHi, false, false);
    }
    // Spill T (D layout: vgpr v -> row v or v+8; lane&15 -> col) to LDS.
    // Hi columns >= 22 are forced to zero (value cndmask, store unconditional).
    const int mBase = (lane < 16) ? 0 : 8;
    const int col = halfM;
    __syncthreads();  // previous consumers of T are done
#pragma unroll
    for (int v = 0; v < 8; ++v) {
        T[(mBase + v) * TSTRIDE + col] = accLo[v];
        T[(mBase + v) * TSTRIDE + 16 + col] = (col < HALO - 16) ? accHi[v] : 0.0f;
    }
    __syncthreads();
    // Stage 2: S = T * Mh, Mh[j][c] = 1 iff c <= j <= c+6 (false for j >= 22)
    v8f accS = {};
#pragma unroll
    for (int q = 0; q < 6; ++q) {
        const int k0 = 4 * q + kOff;
        v2f a2;
        a2.x = T[halfM * TSTRIDE + k0];
        a2.y = T[halfM * TSTRIDE + k0 + 1];
        v2f b2;
        b2.x = (k0 >= halfM && k0 <= halfM + 6) ? 1.0f : 0.0f;
        b2.y = (k0 + 1 >= halfM && k0 + 1 <= halfM + 6) ? 1.0f : 0.0f;
        accS = __builtin_amdgcn_wmma_f32_16x16x4_f32(false, a2, false, b2,
                                                     (short)0, accS, false, false);
    }
    return accS;
}

// ---------------------------------------------------------------------------
// Kernel 2: per 16x16 tile -> pixel_w = (S2 - S1^2/49)/48, accumulate
// Sb[b] += sum(pixel_w * residual). One wave per tile, 8 waves per block.
// Residual reads hit L2 (33.5 MB resident in 192 MB L2).
// ---------------------------------------------------------------------------
__global__ __launch_bounds__(256) void ldl_boxvar_kernel(
    const float* __restrict__ res, float* __restrict__ Sb) {
    __shared__ float Rt[WAVES][RROWS * RSTRIDE];
    __shared__ float Tt[WAVES][TILE * TSTRIDE];
    const int wave = threadIdx.x >> 5;
    const int lane = threadIdx.x & 31;
    const int tileId = blockIdx.x * WAVES + wave;
    const int b = tileId >> 10;        // / TILES_PER_IMG
    const int t = tileId & 1023;
    const int ty = t >> 5, tx = t & 31;
    const int y0 = ty * TILE - PAD, x0 = tx * TILE - PAD;
    const float* rb = res + (size_t)b * HW;
    float* R = Rt[wave];
    float* T = Tt[wave];

    // Load 22x22 halo tile (reflect padding); zero-fill pad region 24x32.
    // Reflect indices stay in-range even for pad rows/cols, so the global
    // load is unconditional and the pad is a cndmask on the value.
#pragma unroll
    for (int idx = lane; idx < RROWS * RCOLS; idx += 32) {
        const int r = idx >> 5, c = idx & 31;
        const float v = rb[reflect_idx(y0 + r) * W + reflect_idx(x0 + c)];
        R[r * RSTRIDE + c] = (r < HALO && c < HALO) ? v : 0.0f;
    }
    __syncthreads();

    const v8f s1 = box7x7_wmma(R, T, false, lane);
    const v8f s2 = box7x7_wmma(R, T, true, lane);

    const int mBase = (lane < 16) ? 0 : 8;
    const int col = lane & 15;
    float partial = 0.0f;
#pragma unroll
    for (int v = 0; v < 8; ++v) {
        const int row = mBase + v;
        const float a = s1[v], bq = s2[v];
        const float pw = (bq - a * a * (1.0f / 49.0f)) * (1.0f / 48.0f);
        partial += pw * R[(row + PAD) * RSTRIDE + (col + PAD)];
    }
    partial = wave_sum32(partial);
    if (lane == 0) atomicAdd(&Sb[b], partial);
}

// ---------------------------------------------------------------------------
// Kernel 3: finalize -> scalar loss
// ---------------------------------------------------------------------------
__global__ void ldl_finalize_kernel(const float* sumB, const float* sumsqB,
                                    const float* Sb, float* out) {
    if (threadIdx.x == 0) {
        const float N = (float)HW;
        float total = 0.0f;
        for (int b = 0; b < BATCH; ++b) {
            const float s = sumB[b];
            const float var = (sumsqB[b] - s * s / N) / (N - 1.0f);
            const float pw = powf(var, 0.2f);
            total += pw * Sb[b];
        }
        out[0] = total * (1.0f / (float)((size_t)BATCH * CH * HW));  // * LOSS_WEIGHT(1)
    }
}

extern "C" void kernel_launch(void* const* d_in, const int* in_sizes, int n_in,
                              void* d_out, int out_size, void* d_ws, size_t ws_size,
                              hipStream_t stream) {
    const float* net = (const float*)d_in[0];
    const float* gt = (const float*)d_in[1];
    float* ws = (float*)d_ws;
    float* sumB = ws;           // 32
    float* sumsqB = ws + 32;    // 32
    float* Sb = ws + 64;        // 32
    float* res = ws + 256;      // 32*512*512 floats = 33.55 MB
    float* out = (float*)d_out;

    ldl_init_kernel<<<1, 128, 0, stream>>>(ws);
    ldl_residual_kernel<<<dim3(HW / 256, BATCH), 256, 0, stream>>>(net, gt, res,
                                                                  sumB, sumsqB);
    ldl_boxvar_kernel<<<NTILES / WAVES, 256, 0, stream>>>(res, Sb);
    ldl_finalize_kernel<<<1, 32, 0, stream>>>(sumB, sumsqB, Sb, out);
}